// BlockBucket_35304631173763
// MI455X (gfx1250) — compile-verified
//
#include <hip/hip_runtime.h>

// ---------------------------------------------------------------------------
// Problem constants (from the reference)
// ---------------------------------------------------------------------------
#define CCH   64          // channels
#define HW    4096        // 64*64 pixels
#define NT    72          // bucket types
#define WROW  577         // C*K*K + 1 (576 weights + 1 bias) per out channel
#define EMBD  36928       // 64 * 577
#define PATK  576         // GEMM K dim
#define KCHUNK 64         // K rows staged per TDM transfer
#define NCHUNK (PATK / KCHUNK)   // 9

typedef float v2f __attribute__((ext_vector_type(2)));
typedef float v8f __attribute__((ext_vector_type(8)));
typedef unsigned int u32x4 __attribute__((ext_vector_type(4)));
typedef int i32x4 __attribute__((ext_vector_type(4)));
typedef int i32x8 __attribute__((ext_vector_type(8)));

// TDM builtin arity differs between toolchains; the gfx1250 TDM header ships
// only with the 6-arg toolchain (therock-10.0 HIP headers).
#if __has_include(<hip/amd_detail/amd_gfx1250_TDM.h>)
#define TDM_SIX_ARGS 1
#else
#define TDM_SIX_ARGS 0
#endif

// group1 word0: workgroup_mask=0 | data_size(4B: code 2)<<16 | pad cfg
#define TDM_CFG_PLAIN 0x00020000u
// + pad_enable(bit20) | pad_interval=5 (64 DWORDs, bits24:22) | pad_amount=1 (2 DWORDs, bits31:25)
#define TDM_CFG_PAD66 (0x00020000u | (1u << 20) | (5u << 22) | (1u << 25))

__device__ __forceinline__ unsigned lds_off(const void* p) {
    // Generic pointer to LDS: low 32 bits are the in-LDS byte address.
    return (unsigned)(unsigned long long)p;
}

// 2D tile load Global -> LDS via the Tensor Data Mover.
// tile_d0 = contiguous elements per row, tile_d1 = rows, stride0 in elements.
__device__ __forceinline__ void tdm_load_2d(unsigned lds_addr,
                                            unsigned long long gaddr,
                                            unsigned tile_d0, unsigned tile_d1,
                                            unsigned stride0_elems,
                                            unsigned w0cfg) {
    const unsigned td0 = 0x7FFFFFFFu;   // huge tensor dims: disable OOB clipping
    const unsigned td1 = 0x7FFFFFFFu;
    u32x4 g0;
    g0[0] = 1u;                                            // count=1 (valid user D#)
    g0[1] = lds_addr;                                      // LDS byte address
    g0[2] = (unsigned)(gaddr & 0xFFFFFFFFull);             // global_addr[31:0]
    g0[3] = (unsigned)((gaddr >> 32) & 0x1FFFFFFull)       // global_addr[56:32]
          | (2u << 30);                                    // type=2 ("image")
    i32x8 g1;
    g1[0] = (int)w0cfg;
    g1[1] = (int)((td0 & 0xFFFFu) << 16);                              // tensor_dim0[15:0]
    g1[2] = (int)(((td0 >> 16) & 0xFFFFu) | ((td1 & 0xFFFFu) << 16));  // dim0 hi | dim1 lo
    g1[3] = (int)(((td1 >> 16) & 0xFFFFu) | ((tile_d0 & 0xFFFFu) << 16)); // dim1 hi | tile_dim0
    g1[4] = (int)(tile_d1 & 0xFFFFu);                                  // tile_dim1, tile_dim2=0
    g1[5] = (int)stride0_elems;                                        // tensor_dim0_stride[31:0]
    g1[6] = 0;                                                         // stride0 hi | stride1 lo
    g1[7] = 0;
    i32x4 z4 = {0, 0, 0, 0};
#if TDM_SIX_ARGS
    i32x8 z8 = {0, 0, 0, 0, 0, 0, 0, 0};
    __builtin_amdgcn_tensor_load_to_lds(g0, g1, z4, z4, z8, 0);
#else
    __builtin_amdgcn_tensor_load_to_lds(g0, g1, z4, z4, 0);
#endif
}

// ---------------------------------------------------------------------------
// Bucket sort of pixels (histogram -> exclusive scan -> scatter + rank)
// ---------------------------------------------------------------------------
__global__ __launch_bounds__(256) void bucket_init(int* counts, int* cursor) {
    int i = threadIdx.x;
    if (i < NT) { counts[i] = 0; cursor[i] = 0; }
}

__global__ __launch_bounds__(256) void bucket_count(const int* __restrict__ buckets,
                                                    int* __restrict__ counts) {
    int p = blockIdx.x * blockDim.x + threadIdx.x;
    if (p < HW) atomicAdd(&counts[buckets[p]], 1);
}

__global__ __launch_bounds__(32) void bucket_scan(const int* __restrict__ counts,
                                                  int* __restrict__ offsets) {
    if (threadIdx.x == 0) {
        int s = 0;
        for (int t = 0; t < NT; ++t) { offsets[t] = s; s += counts[t]; }
        offsets[NT] = s;
    }
}

__global__ __launch_bounds__(256) void bucket_scatter(const int* __restrict__ buckets,
                                                      const int* __restrict__ offsets,
                                                      int* __restrict__ cursor,
                                                      int* __restrict__ pixlist,
                                                      int* __restrict__ rank) {
    int p = blockIdx.x * blockDim.x + threadIdx.x;
    if (p >= HW) return;
    int t = buckets[p];
    int pos = atomicAdd(&cursor[t], 1);
    int s = offsets[t] + pos;
    pixlist[s] = p;
    rank[p] = s;
}

// ---------------------------------------------------------------------------
// Grouped 3x3 conv (groups=4, zero pad=1) + bias + ReLU
// ---------------------------------------------------------------------------
__global__ __launch_bounds__(256) void gconv3x3_relu(const float* __restrict__ x,
                                                     const float* __restrict__ w,
                                                     const float* __restrict__ b,
                                                     float* __restrict__ out) {
    int idx = blockIdx.x * blockDim.x + threadIdx.x;
    if (idx >= CCH * HW) return;
    int pixel = idx & (HW - 1);
    int o     = idx >> 12;
    int g     = o >> 4;
    int y = pixel >> 6, xx = pixel & 63;
    float acc = b[o];
    const float* wo = w + o * 16 * 9;
    for (int ci = 0; ci < 16; ++ci) {
        const float* xc = x + (((size_t)(g * 16 + ci)) << 12);
        const float* wc = wo + ci * 9;
        #pragma unroll
        for (int ky = 0; ky < 3; ++ky) {
            int yy = y + ky - 1;
            if (yy < 0 || yy > 63) continue;
            #pragma unroll
            for (int kx = 0; kx < 3; ++kx) {
                int xc2 = xx + kx - 1;
                if (xc2 < 0 || xc2 > 63) continue;
                acc += xc[(yy << 6) + xc2] * wc[ky * 3 + kx];
            }
        }
    }
    out[idx] = fmaxf(acc, 0.0f);
}

// ---------------------------------------------------------------------------
// Build patch matrix in BUCKET-SORTED column order:
//   P[row][rank[pixel]] = reflect-padded src, row = c*9 + (ky*3+kx)
// Sorted columns make every 16-pixel bucket tile a contiguous 2D tile -> TDM.
// ---------------------------------------------------------------------------
__global__ __launch_bounds__(256) void build_patches(const float* __restrict__ src,
                                                     const int* __restrict__ rank,
                                                     float* __restrict__ P) {
    int idx = blockIdx.x * blockDim.x + threadIdx.x;
    if (idx >= PATK * HW) return;
    int pixel = idx & (HW - 1);
    int row   = idx >> 12;      // c*9 + tap
    int c   = row / 9;
    int tap = row - c * 9;
    int ky = tap / 3, kx = tap - ky * 3;
    int y = pixel >> 6, x = pixel & 63;
    int ry = y + ky - 1; ry = (ry < 0) ? -ry : ((ry > 63) ? 126 - ry : ry);
    int rx = x + kx - 1; rx = (rx < 0) ? -rx : ((rx > 63) ? 126 - rx : rx);
    P[(((size_t)row) << 12) + rank[pixel]] = src[(((size_t)c) << 12) + (ry << 6) + rx];
}

// ---------------------------------------------------------------------------
// Bucket-sorted local conv: TDM double-buffered LDS staging + F32 WMMA.
// grid = (NT, HW/16 tiles); block = 128 threads = 4 waves.
// Wave w computes out-channel tile [16w,16w+16) x 16 sorted pixels.
//   A (16x4 f32): lane<16 -> K=0,1 ; lane>=16 -> K=2,3 ; M=lane&15
//   B (4x16 f32): same K split ; N=lane&15
//   C/D:          VGPR i -> M = i + 8*(lane>>4), N = lane&15
// ---------------------------------------------------------------------------
__global__ __launch_bounds__(128) void localconv_wmma(const float* __restrict__ P,
                                                      const float* __restrict__ emb,
                                                      const int* __restrict__ pixlist,
                                                      const int* __restrict__ offsets,
                                                      float* __restrict__ L) {
    const int t     = blockIdx.x;
    const int tile  = blockIdx.y;
    const int start = offsets[t];
    const int cnt   = offsets[t + 1] - start;
    if (tile * 16 >= cnt) return;           // uniform across block
    const int colbase = start + tile * 16;  // sorted column of this pixel tile

    __shared__ float Wbuf[2][64][66];       // 66 = TDM pad -> conflict-free A reads
    __shared__ float Bbuf[2][KCHUNK][16];
    __shared__ int   pix[16];

    const int lane = threadIdx.x & 31;
    const int wave = threadIdx.x >> 5;
    if (threadIdx.x < 16) {
        int idx = tile * 16 + threadIdx.x;
        pix[threadIdx.x] = pixlist[start + ((idx < cnt) ? idx : tile * 16)];
    }

    const int col   = lane & 15;            // pixel column / A row (M)
    const int krow  = (lane >> 4) << 1;     // 0 or 2
    const int obase = wave * 16;

    const unsigned long long embT = (unsigned long long)(uintptr_t)emb
                                  + (unsigned long long)t * (EMBD * 4ull);
    const unsigned long long pB   = (unsigned long long)(uintptr_t)P
                                  + (unsigned long long)colbase * 4ull;

    if (wave == 0) {  // prologue: chunk 0 into buffer 0
        tdm_load_2d(lds_off(&Wbuf[0][0][0]), embT, KCHUNK, 64, WROW, TDM_CFG_PAD66);
        tdm_load_2d(lds_off(&Bbuf[0][0][0]), pB, 16, KCHUNK, HW, TDM_CFG_PLAIN);
    }

    v8f acc = {};
    for (int c = 0; c < NCHUNK; ++c) {
        const int buf = c & 1;
        if (wave == 0) {
            if (c + 1 < NCHUNK) {
                const unsigned long long gW = embT + (unsigned long long)(c + 1) * (KCHUNK * 4ull);
                const unsigned long long gB = pB + (unsigned long long)(c + 1) * (KCHUNK * (unsigned long long)HW * 4ull);
                tdm_load_2d(lds_off(&Wbuf[buf ^ 1][0][0]), gW, KCHUNK, 64, WROW, TDM_CFG_PAD66);
                tdm_load_2d(lds_off(&Bbuf[buf ^ 1][0][0]), gB, 16, KCHUNK, HW, TDM_CFG_PLAIN);
                __builtin_amdgcn_s_wait_tensorcnt(2);   // chunk c landed (in-order)
            } else {
                __builtin_amdgcn_s_wait_tensorcnt(0);
            }
        }
        __syncthreads();                    // LDS chunk c visible to all waves
        #pragma unroll
        for (int ks = 0; ks < KCHUNK / 4; ++ks) {
            const int kq = ks * 4 + krow;
            v2f a = *(const v2f*)&Wbuf[buf][obase + col][kq];   // ds_load_b64
            v2f b;
            b.x = Bbuf[buf][kq][col];
            b.y = Bbuf[buf][kq + 1][col];
            acc = __builtin_amdgcn_wmma_f32_16x16x4_f32(
                /*neg_a=*/false, a, /*neg_b=*/false, b,
                /*c_mod=*/(short)0, acc, /*reuse_a=*/false, /*reuse_b=*/false);
        }
        __syncthreads();                    // free buffer before TDM overwrites it
    }

    const int  mhalf = (lane >> 4) << 3;    // 0 or 8
    const bool valid = (tile * 16 + col) < cnt;
    const int  mypix = pix[col];
    const float* wt  = emb + (size_t)t * EMBD;
    #pragma unroll
    for (int i = 0; i < 8; ++i) {
        int o = obase + mhalf + i;
        float v = acc[i] + wt[(size_t)o * WROW + PATK];   // per-channel bias
        if (valid) L[(((size_t)o) << 12) + mypix] = v;    // scatter back to pixel order
    }
}

// ---------------------------------------------------------------------------
// out = relu( w2(1x1) * relu(L) + b2 + xin )   (residual close of eres block)
// ---------------------------------------------------------------------------
__global__ __launch_bounds__(256) void conv1x1_res_relu(const float* __restrict__ L,
                                                        const float* __restrict__ w2,
                                                        const float* __restrict__ b2,
                                                        const float* __restrict__ xin,
                                                        float* __restrict__ out) {
    int idx = blockIdx.x * blockDim.x + threadIdx.x;
    if (idx >= CCH * HW) return;
    int pixel = idx & (HW - 1);
    int o     = idx >> 12;
    float acc = b2[o];
    const float* wo = w2 + o * CCH;
    #pragma unroll 4
    for (int c = 0; c < CCH; ++c)
        acc += fmaxf(L[(((size_t)c) << 12) + pixel], 0.0f) * wo[c];
    out[idx] = fmaxf(acc + xin[idx], 0.0f);
}

// ---------------------------------------------------------------------------
// basic_block: 1x1 conv over implicit concat of up to 4 sources (+bias, ReLU)
// ---------------------------------------------------------------------------
__global__ __launch_bounds__(256) void conv1x1_cat_relu(const float* __restrict__ s0,
                                                        const float* __restrict__ s1,
                                                        const float* __restrict__ s2,
                                                        const float* __restrict__ s3,
                                                        int nsrc,
                                                        const float* __restrict__ w,
                                                        const float* __restrict__ b,
                                                        float* __restrict__ out) {
    int idx = blockIdx.x * blockDim.x + threadIdx.x;
    if (idx >= CCH * HW) return;
    int pixel = idx & (HW - 1);
    int o     = idx >> 12;
    const int cin = nsrc * CCH;
    float acc = b[o];
    const float* wo = w + (size_t)o * cin;
    const float* srcs[4] = {s0, s1, s2, s3};
    for (int s = 0; s < nsrc; ++s) {
        const float* sp = srcs[s];
        const float* wsrc = wo + s * CCH;
        #pragma unroll 4
        for (int c = 0; c < CCH; ++c)
            acc += sp[(((size_t)c) << 12) + pixel] * wsrc[c];
    }
    out[idx] = fmaxf(acc, 0.0f);
}

// ---------------------------------------------------------------------------
// Host-side orchestration (graph-capture safe: only kernel launches on stream)
// ---------------------------------------------------------------------------
extern "C" void kernel_launch(void* const* d_in, const int* in_sizes, int n_in,
                              void* d_out, int out_size, void* d_ws, size_t ws_size,
                              hipStream_t stream) {
    const float* x       = (const float*)d_in[0];
    const int*   buckets = (const int*)d_in[1];
    const float* w1[3]  = {(const float*)d_in[2],  (const float*)d_in[7],  (const float*)d_in[12]};
    const float* bb1[3] = {(const float*)d_in[3],  (const float*)d_in[8],  (const float*)d_in[13]};
    const float* emb[3] = {(const float*)d_in[4],  (const float*)d_in[9],  (const float*)d_in[14]};
    const float* w2[3]  = {(const float*)d_in[5],  (const float*)d_in[10], (const float*)d_in[15]};
    const float* bb2[3] = {(const float*)d_in[6],  (const float*)d_in[11], (const float*)d_in[16]};
    const float* cw[3]  = {(const float*)d_in[17], (const float*)d_in[19], (const float*)d_in[21]};
    const float* cb[3]  = {(const float*)d_in[18], (const float*)d_in[20], (const float*)d_in[22]};

    // Workspace carve-up
    char* ws = (char*)d_ws;
    float* P   = (float*)ws; ws += (size_t)PATK * HW * sizeof(float);   // 9.44 MB (sorted cols)
    float* t1  = (float*)ws; ws += (size_t)CCH  * HW * sizeof(float);
    float* L   = (float*)ws; ws += (size_t)CCH  * HW * sizeof(float);
    float* bo[3];
    for (int i = 0; i < 3; ++i) { bo[i] = (float*)ws; ws += (size_t)CCH * HW * sizeof(float); }
    float* o1  = (float*)ws; ws += (size_t)CCH * HW * sizeof(float);
    float* o2  = (float*)ws; ws += (size_t)CCH * HW * sizeof(float);
    int* counts  = (int*)ws; ws += NT * sizeof(int);
    int* cursor  = (int*)ws; ws += NT * sizeof(int);
    int* offsets = (int*)ws; ws += (NT + 1) * sizeof(int);
    int* pixlist = (int*)ws; ws += HW * sizeof(int);
    int* rank    = (int*)ws; ws += HW * sizeof(int);

    const dim3 blk256(256), blk128(128);
    const int  gPix = (CCH * HW + 255) / 256;          // (o, pixel) kernels
    const int  gPat = (PATK * HW + 255) / 256;

    // --- bucket sort (shared by all three eres blocks) ---
    bucket_init   <<<1, blk256, 0, stream>>>(counts, cursor);
    bucket_count  <<<(HW + 255) / 256, blk256, 0, stream>>>(buckets, counts);
    bucket_scan   <<<1, 32, 0, stream>>>(counts, offsets);
    bucket_scatter<<<(HW + 255) / 256, blk256, 0, stream>>>(buckets, offsets, cursor,
                                                            pixlist, rank);

    const float* blkin[3]  = {x, o1, o2};
    float*       blkout[3] = {o1, o2, (float*)d_out};

    for (int i = 0; i < 3; ++i) {
        // eres block
        gconv3x3_relu <<<gPix, blk256, 0, stream>>>(blkin[i], w1[i], bb1[i], t1);
        build_patches <<<gPat, blk256, 0, stream>>>(t1, rank, P);
        localconv_wmma<<<dim3(NT, HW / 16), blk128, 0, stream>>>(P, emb[i], pixlist, offsets, L);
        conv1x1_res_relu<<<gPix, blk256, 0, stream>>>(L, w2[i], bb2[i], blkin[i], bo[i]);
        // basic block on concat [x, bo[0..i]]
        const float* s1 = bo[0];
        const float* s2 = (i >= 1) ? bo[1] : bo[0];
        const float* s3 = (i >= 2) ? bo[2] : bo[0];
        conv1x1_cat_relu<<<gPix, blk256, 0, stream>>>(x, s1, s2, s3, i + 2,
                                                      cw[i], cb[i], blkout[i]);
    }
}